// TAGNet_70394513981880
// MI455X (gfx1250) — compile-verified
//
#include <hip/hip_runtime.h>
#include <math.h>

// Problem constants (match reference)
#define NN 20000
#define EE 480000
#define BB 4
#define DD 32
#define RR 474
#define LLAYERS 3
#define TT 64
#define KDIM (13 * DD) // 416

typedef __attribute__((ext_vector_type(2))) float v2f;
typedef __attribute__((ext_vector_type(8))) float v8f;

// ---- CDNA5 native no-return fp32 atomics (avoid CAS-loop lowering) ----
__device__ __forceinline__ void atomAddF(float* p, float v) {
  asm volatile("global_atomic_add_f32 %0, %1, off" :: "v"(p), "v"(v) : "memory");
}
__device__ __forceinline__ void atomMaxF(float* p, float v) {
  asm volatile("global_atomic_max_num_f32 %0, %1, off" :: "v"(p), "v"(v) : "memory");
}
__device__ __forceinline__ void atomMinF(float* p, float v) {
  asm volatile("global_atomic_min_num_f32 %0, %1, off" :: "v"(p), "v"(v) : "memory");
}

// query[b,d] = query_weight[r_index[b], d]
__global__ void k_query(const float* __restrict__ qw, const int* __restrict__ r_index,
                        float* __restrict__ query) {
  int i = blockIdx.x * blockDim.x + threadIdx.x;
  if (i < BB * DD) {
    int b = i / DD, d = i % DD;
    query[i] = qw[r_index[b] * DD + d];
  }
}

// hidden0 = boundary (zeros, query scattered at h_index)
__global__ void k_init_hidden(const float* __restrict__ query, const int* __restrict__ h_index,
                              float* __restrict__ hid) {
  int i = blockIdx.x * blockDim.x + threadIdx.x;
  if (i >= BB * NN * DD) return;
  int d = i % DD;
  int n = (i / DD) % NN;
  int b = i / (NN * DD);
  hid[i] = (n == h_index[b]) ? query[b * DD + d] : 0.0f;
}

__global__ void k_zero_deg(float* __restrict__ degf, float* __restrict__ logsum) {
  int i = blockIdx.x * blockDim.x + threadIdx.x;
  if (i < NN) degf[i] = 0.0f;
  if (i == 0) logsum[0] = 0.0f;
}

__global__ void k_deg(const int* __restrict__ dst, float* __restrict__ degf) {
  int e = blockIdx.x * blockDim.x + threadIdx.x;
  if (e < EE) atomAddF(&degf[dst[e]], 1.0f);
}

// deg_total = indeg + 1 (self loop); logv = log(deg_total + 1); logsum += logv
__global__ void k_logdeg(float* __restrict__ degf, float* __restrict__ logv,
                         float* __restrict__ logsum) {
  int n = blockIdx.x * blockDim.x + threadIdx.x;
  if (n >= NN) return;
  float dtot = degf[n] + 1.0f;
  degf[n] = dtot;
  float lv = logf(dtot + 1.0f);
  logv[n] = lv;
  atomAddF(logsum, lv);
}

// PNA scalers: [1, scale, 1/max(scale,1e-2)]
__global__ void k_scalers(const float* __restrict__ logv, const float* __restrict__ logsum,
                          float* __restrict__ scal) {
  int n = blockIdx.x * blockDim.x + threadIdx.x;
  if (n >= NN) return;
  float mean = logsum[0] * (1.0f / (float)NN);
  float s = logv[n] / mean;
  scal[n * 3 + 0] = 1.0f;
  scal[n * 3 + 1] = s;
  scal[n * 3 + 2] = 1.0f / fmaxf(s, 1e-2f);
}

// rel[l,b,r,d] = sum_k query[b,k] * rel_lin_w[l,k, r*D+d]
__global__ void k_rel(const float* __restrict__ query, const float* __restrict__ rlw,
                      float* __restrict__ rel) {
  int i = blockIdx.x * blockDim.x + threadIdx.x;
  if (i >= LLAYERS * BB * RR * DD) return;
  int d = i % DD;
  int r = (i / DD) % RR;
  int b = (i / (DD * RR)) % BB;
  int l = i / (DD * RR * BB);
  float acc = 0.0f;
#pragma unroll
  for (int k = 0; k < DD; ++k)
    acc += query[b * DD + k] * rlw[(size_t)(l * DD + k) * (RR * DD) + r * DD + d];
  rel[i] = acc;
}

// Aggregate init = boundary self-loop message (sum/sumsq/max/min all start at it)
__global__ void k_init_agg(const float* __restrict__ query, const int* __restrict__ h_index,
                           float* __restrict__ sumv, float* __restrict__ sqv,
                           float* __restrict__ mxv, float* __restrict__ mnv) {
  int i = blockIdx.x * blockDim.x + threadIdx.x;
  if (i >= BB * NN * DD) return;
  int d = i % DD;
  int n = (i / DD) % NN;
  int b = i / (NN * DD);
  float v = (n == h_index[b]) ? query[b * DD + d] : 0.0f;
  sumv[i] = v;
  sqv[i] = v * v;
  mxv[i] = v;
  mnv[i] = v;
}

// DistMult message + scatter aggregation (lane = feature dim, one wave32 per edge)
__global__ void k_msg(const float* __restrict__ hid, const float* __restrict__ rel_l,
                      const int* __restrict__ src, const int* __restrict__ dst,
                      const int* __restrict__ etype,
                      float* __restrict__ sumv, float* __restrict__ sqv,
                      float* __restrict__ mxv, float* __restrict__ mnv) {
  int i = blockIdx.x * blockDim.x + threadIdx.x;
  if (i >= BB * EE * DD) return;
  int d = i % DD;
  int e = (i / DD) % EE;
  int b = i / (EE * DD);
  int s = src[e];
  int t = dst[e];
  int ty = etype[e];
  float m = hid[((size_t)b * NN + s) * DD + d] * rel_l[((size_t)b * RR + ty) * DD + d];
  size_t o = ((size_t)b * NN + t) * DD + d;
  atomAddF(&sumv[o], m);
  atomAddF(&sqv[o], m * m);
  atomMaxF(&mxv[o], m);
  atomMinF(&mnv[o], m);
}

// Fused PNA feature build + [16,416]@[416,32] WMMA GEMM + bias + relu + residual.
// One block = 16 nodes (N=20000 is 1250*16, tiles never cross batches).
// 2 waves; wave w owns output columns [16w,16w+16).
__global__ void __launch_bounds__(64)
k_update(const float* __restrict__ hid, const float* __restrict__ sumv,
         const float* __restrict__ sqv, const float* __restrict__ mxv,
         const float* __restrict__ mnv, const float* __restrict__ degf,
         const float* __restrict__ scal, const float* __restrict__ W,
         const float* __restrict__ bias, float* __restrict__ hid_out) {
  __shared__ float Xs[16][KDIM + 1]; // +1 pad: odd stride for LDS banks
  int tile = blockIdx.x; // 0..4999
  int b = tile / (NN / 16);
  int n0 = (tile % (NN / 16)) * 16;
  int tid = threadIdx.x;

  // Build the 16 x 416 feature tile: [hidden | scaler_s * {mean,max,min,std}]
  for (int i = tid; i < 16 * KDIM; i += 64) {
    int row = i / KDIM;
    int col = i % KDIM;
    int n = n0 + row;
    float val;
    if (col < DD) {
      val = hid[((size_t)b * NN + n) * DD + col];
    } else {
      int cc = col - DD;
      int sIdx = cc >> 7;      // scaler block (128 cols each)
      int f = (cc >> 5) & 3;   // feature: mean,max,min,std
      int d = cc & 31;
      size_t o = ((size_t)b * NN + n) * DD + d;
      float dg = degf[n];
      float mean = sumv[o] / dg;
      float feat;
      if (f == 0) feat = mean;
      else if (f == 1) feat = mxv[o];
      else if (f == 2) feat = mnv[o];
      else {
        float var = sqv[o] / dg - mean * mean;
        feat = sqrtf(fmaxf(var, 0.0f));
      }
      val = scal[n * 3 + sIdx] * feat;
    }
    Xs[row][col] = val;
  }
  __syncthreads();

  int wave = tid >> 5;
  int lane = tid & 31;
  int nb = wave * 16; // output column base
  int m = lane & 15;  // A row / B col / D col
  int kg = lane >> 4; // K sub-group

#if defined(__gfx1250__) && __has_builtin(__builtin_amdgcn_wmma_f32_16x16x4_f32)
  // f32 WMMA: A 16x4 (lane m, K={2kg,2kg+1} in v[0:1]), B 4x16 mirrored.
  v8f c = {};
  for (int k0 = 0; k0 < KDIM; k0 += 4) {
    v2f a, bb;
    a.x = Xs[m][k0 + 2 * kg + 0];
    a.y = Xs[m][k0 + 2 * kg + 1];
    bb.x = W[(size_t)(k0 + 2 * kg + 0) * DD + nb + m];
    bb.y = W[(size_t)(k0 + 2 * kg + 1) * DD + nb + m];
    c = __builtin_amdgcn_wmma_f32_16x16x4_f32(false, a, false, bb, (short)0, c,
                                              false, false);
  }
#pragma unroll
  for (int r = 0; r < 8; ++r) {
    int mrow = r + 8 * kg; // C/D layout: lanes 16-31 hold M=8..15
    int n = n0 + mrow;
    size_t o = ((size_t)b * NN + n) * DD + nb + m;
    float v = c[r] + bias[nb + m];
    hid_out[o] = fmaxf(v, 0.0f) + hid[o];
  }
#else
  // Scalar fallback (same mapping)
  for (int r = 0; r < 8; ++r) {
    int mrow = r + 8 * kg;
    float acc = 0.0f;
    for (int k = 0; k < KDIM; ++k) acc += Xs[mrow][k] * W[(size_t)k * DD + nb + m];
    int n = n0 + mrow;
    size_t o = ((size_t)b * NN + n) * DD + nb + m;
    float v = acc + bias[nb + m];
    hid_out[o] = fmaxf(v, 0.0f) + hid[o];
  }
#endif
}

// Final MLP scoring: one block per (b,t); 64 threads = 2D hidden width
__global__ void __launch_bounds__(64)
k_score(const float* __restrict__ hid, const float* __restrict__ query,
        const int* __restrict__ t_index, const float* __restrict__ w1,
        const float* __restrict__ b1, const float* __restrict__ w2,
        const float* __restrict__ b2, float* __restrict__ out) {
  __shared__ float feat[2 * DD];
  __shared__ float red[2 * DD];
  int bt = blockIdx.x;
  int b = bt / TT;
  int t = bt % TT;
  int j = threadIdx.x;
  int node = t_index[b * TT + t];
  feat[j] = (j < DD) ? hid[((size_t)b * NN + node) * DD + j] : query[b * DD + (j - DD)];
  __syncthreads();
  float acc = b1[j];
#pragma unroll
  for (int k = 0; k < 2 * DD; ++k) acc += feat[k] * w1[k * (2 * DD) + j];
  red[j] = fmaxf(acc, 0.0f) * w2[j];
  __syncthreads();
  for (int s = 32; s > 0; s >>= 1) {
    if (j < s) red[j] += red[j + s];
    __syncthreads();
  }
  if (j == 0) out[bt] = red[0] + b2[0];
}

extern "C" void kernel_launch(void* const* d_in, const int* in_sizes, int n_in,
                              void* d_out, int out_size, void* d_ws, size_t ws_size,
                              hipStream_t stream) {
  const int* edge_index = (const int*)d_in[0];
  const int* src = edge_index;
  const int* dst = edge_index + EE;
  const int* etype = (const int*)d_in[1];
  const int* h_index = (const int*)d_in[2];
  const int* r_index = (const int*)d_in[3];
  const int* t_index = (const int*)d_in[4];
  const float* query_weight = (const float*)d_in[5];
  const float* rel_lin_w = (const float*)d_in[6];
  const float* layer_w = (const float*)d_in[7];
  const float* layer_b = (const float*)d_in[8];
  const float* mlp_w1 = (const float*)d_in[9];
  const float* mlp_b1 = (const float*)d_in[10];
  const float* mlp_w2 = (const float*)d_in[11];
  const float* mlp_b2 = (const float*)d_in[12];
  float* out = (float*)d_out;

  float* ws = (float*)d_ws;
  size_t off = 0;
  float* query = ws + off;  off += BB * DD;
  float* rel = ws + off;    off += (size_t)LLAYERS * BB * RR * DD;
  float* hidA = ws + off;   off += (size_t)BB * NN * DD;
  float* hidB = ws + off;   off += (size_t)BB * NN * DD;
  float* sumv = ws + off;   off += (size_t)BB * NN * DD;
  float* sqv = ws + off;    off += (size_t)BB * NN * DD;
  float* mxv = ws + off;    off += (size_t)BB * NN * DD;
  float* mnv = ws + off;    off += (size_t)BB * NN * DD;
  float* degf = ws + off;   off += NN;
  float* logv = ws + off;   off += NN;
  float* scal = ws + off;   off += 3 * NN;
  float* logsum = ws + off; off += 4;

  const int TPB = 256;
  // Precompute: query, boundary hidden, degree scalers, relation tables
  k_query<<<1, 128, 0, stream>>>(query_weight, r_index, query);
  k_init_hidden<<<(BB * NN * DD) / TPB, TPB, 0, stream>>>(query, h_index, hidA);
  k_zero_deg<<<(NN + TPB - 1) / TPB, TPB, 0, stream>>>(degf, logsum);
  k_deg<<<(EE + TPB - 1) / TPB, TPB, 0, stream>>>(dst, degf);
  k_logdeg<<<(NN + TPB - 1) / TPB, TPB, 0, stream>>>(degf, logv, logsum);
  k_scalers<<<(NN + TPB - 1) / TPB, TPB, 0, stream>>>(logv, logsum, scal);
  k_rel<<<(LLAYERS * BB * RR * DD + TPB - 1) / TPB, TPB, 0, stream>>>(query, rel_lin_w, rel);

  float* cur = hidA;
  float* nxt = hidB;
  for (int l = 0; l < LLAYERS; ++l) {
    k_init_agg<<<(BB * NN * DD) / TPB, TPB, 0, stream>>>(query, h_index, sumv, sqv, mxv, mnv);
    k_msg<<<(BB * EE * DD) / TPB, TPB, 0, stream>>>(
        cur, rel + (size_t)l * BB * RR * DD, src, dst, etype, sumv, sqv, mxv, mnv);
    k_update<<<(BB * NN) / 16, 64, 0, stream>>>(
        cur, sumv, sqv, mxv, mnv, degf, scal,
        layer_w + (size_t)l * KDIM * DD, layer_b + (size_t)l * DD, nxt);
    float* tmp = cur; cur = nxt; nxt = tmp;
  }

  k_score<<<BB * TT, 64, 0, stream>>>(cur, query, t_index, mlp_w1, mlp_b1, mlp_w2,
                                      mlp_b2, out);
}